// PupilCenterRegressionHead_41540923687404
// MI455X (gfx1250) — compile-verified
//
#include <hip/hip_runtime.h>
#include <math.h>

#define B_   32
#define C_   256
#define H_   64
#define W_   64
#define MIP_ 8
#define HID_ 256
#define OUT_ 14

typedef __attribute__((ext_vector_type(2))) float v2f;
typedef __attribute__((ext_vector_type(8))) float v8f;

__device__ __forceinline__ v8f wmma_f32_16x16x4(v2f a, v2f b, v8f c) {
  // 8 args: (neg_a, A, neg_b, B, c_mod, C, reuse_a, reuse_b)
  return __builtin_amdgcn_wmma_f32_16x16x4_f32(false, a, false, b, (short)0, c,
                                               false, false);
}

__device__ __forceinline__ float softplus_f(float x) {
  return (x > 20.f) ? x : log1pf(expf(x));
}

// ---------------------------------------------------------------------------
// Kernel 1: per-(b,c) 64x64 tile -> row means (xh) and column means (xw).
// float4 (b128) streaming loads; padded-LDS (pitch 65) so both reduction
// directions are bank-conflict-free.
// ---------------------------------------------------------------------------
__global__ void pool_kernel(const float* __restrict__ x,
                            float* __restrict__ xh, float* __restrict__ xw) {
  __shared__ float tile[64 * 65];
  const int bc  = blockIdx.x;
  const int tid = threadIdx.x;
  const float4* xp4 = (const float4*)(x + (size_t)bc * (H_ * W_));
#pragma unroll
  for (int it = 0; it < 4; ++it) {
    int idx4 = it * 256 + tid;       // 0..1023 float4's
    float4 v = xp4[idx4];
    int h = idx4 >> 4;               // 16 float4 per row
    int w = (idx4 & 15) * 4;
    tile[h * 65 + w + 0] = v.x;
    tile[h * 65 + w + 1] = v.y;
    tile[h * 65 + w + 2] = v.z;
    tile[h * 65 + w + 3] = v.w;
  }
  __syncthreads();
  if (tid < 64) {
    float s = 0.f;
#pragma unroll
    for (int i = 0; i < 64; ++i) s += tile[tid * 65 + i];
    xh[(size_t)bc * 64 + tid] = s * (1.0f / 64.0f);
  } else if (tid < 128) {
    int w = tid - 64;
    float s = 0.f;
#pragma unroll
    for (int i = 0; i < 64; ++i) s += tile[i * 65 + w];
    xw[(size_t)bc * 64 + w] = s * (1.0f / 64.0f);
  }
}

// ---------------------------------------------------------------------------
// Kernel 2: coordinate attention MLP, one block per batch, all GEMMs on
// V_WMMA_F32_16X16X4_F32 (f32 in, f32 accumulate). ca_w1 is staged into LDS
// zero-padded to 16 rows so A-fragment loads are branch-free.
// ---------------------------------------------------------------------------
__global__ void attn_kernel(const float* __restrict__ xh,
                            const float* __restrict__ xw,
                            const float* __restrict__ ca_w1,
                            const float* __restrict__ ca_g,
                            const float* __restrict__ ca_b,
                            const float* __restrict__ ca_m,
                            const float* __restrict__ ca_v,
                            const float* __restrict__ ca_wh,
                            const float* __restrict__ ca_ww,
                            float* __restrict__ a_h, float* __restrict__ a_w) {
  __shared__ float w1s[16][C_];   // ca_w1 zero-padded 8 -> 16 rows
  __shared__ float yh[MIP_][128];
  const int b    = blockIdx.x;
  const int tid  = threadIdx.x;
  const int wave = tid >> 5;
  const int lane = tid & 31;
  const int ll   = lane & 15;  // low lane index (n or m within tile)
  const int hi   = lane >> 4;  // upper half-wave flag
  const float* xhb = xh + (size_t)b * C_ * 64;
  const float* xwb = xw + (size_t)b * C_ * 64;

  // stage padded A matrix into LDS (branch-free WMMA operand loads later)
  for (int i = tid; i < 16 * C_; i += 256) {
    int m = i >> 8, k = i & (C_ - 1);
    w1s[m][k] = (m < MIP_) ? ca_w1[m * C_ + k] : 0.f;
  }
  __syncthreads();

  // ---- Stage 1: y2(16x128) = pad(ca_w1)(16x256) @ y(256x128); wave = n-tile
  {
    const int n = wave * 16 + ll;                 // 0..127
    const float* src = (n >= 64) ? (xwb + (n - 64)) : (xhb + n);
    v8f acc = {};
    for (int kb = 0; kb < C_; kb += 4) {
      int k0 = kb + 2 * hi;
      v2f a, bv;
      a[0]  = w1s[ll][k0];
      a[1]  = w1s[ll][k0 + 1];
      bv[0] = src[(size_t)k0 * 64];
      bv[1] = src[(size_t)(k0 + 1) * 64];
      acc = wmma_f32_16x16x4(a, bv, acc);
    }
    if (hi == 0) {  // rows m=0..7 live in lanes 0..15, d[r] = row r
#pragma unroll
      for (int r = 0; r < MIP_; ++r) yh[r][wave * 16 + ll] = acc[r];
    }
  }
  __syncthreads();

  // ---- BN (axis=1, eps=1e-5) + hard-swish, in LDS
  for (int i = tid; i < MIP_ * 128; i += 256) {
    int mc = i >> 7, l = i & 127;
    float sc  = ca_g[mc] * rsqrtf(ca_v[mc] + 1e-5f);
    float val = (yh[mc][l] - ca_m[mc]) * sc + ca_b[mc];
    yh[mc][l] = val * fminf(fmaxf(val + 3.f, 0.f), 6.f) * (1.f / 6.f);
  }
  __syncthreads();

  // ---- Stage 2: a_h = sigmoid(ca_wh(256x8) @ yh[:, :64]),
  //               a_w = sigmoid(ca_ww(256x8) @ yh[:, 64:]); 128 16x16 tiles
  const size_t base = (size_t)b * C_ * 64;
  for (int t = wave; t < 128; t += 8) {
    int sel   = t >> 6;   // 0: a_h, 1: a_w  (wave-uniform)
    int tt    = t & 63;
    int mtile = tt >> 2;  // 0..15 over C
    int ntile = tt & 3;   // 0..3 over 64 cols
    const float* Wm = sel ? ca_ww : ca_wh;
    int crow = mtile * 16 + ll;
    int ncol = ntile * 16 + ll;
    v8f acc = {};
#pragma unroll
    for (int kb = 0; kb < MIP_; kb += 4) {
      int k0 = kb + 2 * hi;
      v2f a, bv;
      a[0]  = Wm[crow * MIP_ + k0];
      a[1]  = Wm[crow * MIP_ + k0 + 1];
      bv[0] = yh[k0][sel * 64 + ncol];
      bv[1] = yh[k0 + 1][sel * 64 + ncol];
      acc = wmma_f32_16x16x4(a, bv, acc);
    }
    float* dst = sel ? a_w : a_h;
#pragma unroll
    for (int r = 0; r < 8; ++r) {
      int c = mtile * 16 + r + 8 * hi;
      float s = 1.f / (1.f + expf(-acc[r]));
      dst[base + (size_t)c * 64 + ncol] = s;
    }
  }
}

// ---------------------------------------------------------------------------
// Kernel 3: p[b,c] = (1/4096) * sum_{h,w} x[h,w] * a_h[h] * a_w[w]
// Second streaming pass over x with b128 loads (expected to hit the 192MB L2).
// Each thread owns a fixed 4-column group; a_w is read once per thread.
// ---------------------------------------------------------------------------
__global__ void wpool_kernel(const float* __restrict__ x,
                             const float* __restrict__ a_h,
                             const float* __restrict__ a_w,
                             float* __restrict__ p) {
  __shared__ float red[256];
  const int bc  = blockIdx.x;
  const int tid = threadIdx.x;
  const float4* xp4 = (const float4*)(x + (size_t)bc * 4096);
  const float*  ah  = a_h + (size_t)bc * 64;
  const float4* aw4 = (const float4*)(a_w + (size_t)bc * 64);
  const int cg = tid & 15;   // column group (4 cols)
  const int h0 = tid >> 4;   // 0..15
  const float4 aw = aw4[cg];
  float acc = 0.f;
#pragma unroll
  for (int it = 0; it < 4; ++it) {
    int h = h0 + 16 * it;
    float4 v = xp4[h * 16 + cg];
    float partial = v.x * aw.x + v.y * aw.y + v.z * aw.z + v.w * aw.w;
    acc += partial * ah[h];
  }
  red[tid] = acc;
  __syncthreads();
  for (int s = 128; s > 0; s >>= 1) {
    if (tid < s) red[tid] += red[tid + s];
    __syncthreads();
  }
  if (tid == 0) p[bc] = red[0] * (1.f / 4096.f);
}

// ---------------------------------------------------------------------------
// Kernel 4: hdd = relu(BN(p(32x256) @ fc1_w^T(256x256) + fc1_b)), WMMA tiles.
// ---------------------------------------------------------------------------
__global__ void fc1_kernel(const float* __restrict__ p,
                           const float* __restrict__ w1,
                           const float* __restrict__ b1,
                           const float* __restrict__ g,
                           const float* __restrict__ bb,
                           const float* __restrict__ mm,
                           const float* __restrict__ vv,
                           float* __restrict__ hdd) {
  const int tid  = threadIdx.x;
  const int wave = tid >> 5;
  const int lane = tid & 31;
  const int ll   = lane & 15;
  const int hi   = lane >> 4;
  for (int t = wave * 4; t < wave * 4 + 4; ++t) {  // 32 tiles, 4 per wave
    int mtile = t >> 4, ntile = t & 15;
    int row  = mtile * 16 + ll;   // batch row (A)
    int ncol = ntile * 16 + ll;   // hidden col (B)
    v8f acc = {};
    for (int kb = 0; kb < C_; kb += 4) {
      int k0 = kb + 2 * hi;
      v2f a, bv;
      a[0]  = p[row * C_ + k0];
      a[1]  = p[row * C_ + k0 + 1];
      bv[0] = w1[ncol * C_ + k0];
      bv[1] = w1[ncol * C_ + k0 + 1];
      acc = wmma_f32_16x16x4(a, bv, acc);
    }
#pragma unroll
    for (int r = 0; r < 8; ++r) {
      int m = mtile * 16 + r + 8 * hi;  // batch 0..31
      int j = ntile * 16 + ll;
      float val = acc[r] + b1[j];
      float sc  = g[j] * rsqrtf(vv[j] + 1e-5f);
      val = (val - mm[j]) * sc + bb[j];
      hdd[m * HID_ + j] = fmaxf(val, 0.f);
    }
  }
}

// ---------------------------------------------------------------------------
// Kernel 5: output head + conic geometry, one thread per (b, e).
// ---------------------------------------------------------------------------
__global__ void head_kernel(const float* __restrict__ hdd,
                            const float* __restrict__ wout,
                            const float* __restrict__ bout,
                            const float* __restrict__ Kmat,
                            const float* __restrict__ iris_r,
                            float* __restrict__ out) {
  const int t = threadIdx.x;
  if (t >= B_ * 2) return;
  const int b = t >> 1, e = t & 1;
  const float* hb = hdd + b * HID_;
  float raw[7];
  for (int j = 0; j < 7; ++j) {
    const float* wr = wout + (e * 7 + j) * HID_;
    float s = 0.f;
    for (int c = 0; c < HID_; ++c) s += hb[c] * wr[c];
    raw[j] = s + bout[e * 7 + j];
  }
  const float cx = raw[0], cy = raw[1];
  const float ea = softplus_f(raw[2]) + 1e-6f;
  const float eb = softplus_f(raw[3]) + 1e-6f;
  const float nrm = sqrtf(raw[4] * raw[4] + raw[5] * raw[5]) + 1e-8f;
  const float cth = raw[4] / nrm, sth = raw[5] / nrm;
  const float delta = 0.3f * tanhf(raw[6]);

  const int be = b * 2 + e;
  out[be * 6 + 0] = cx;  out[be * 6 + 1] = cy;
  out[be * 6 + 2] = ea;  out[be * 6 + 3] = eb;
  out[be * 6 + 4] = cth; out[be * 6 + 5] = sth;
  out[384 + be] = delta;

  const float theta = atan2f(sth, cth);
  const float ct = cosf(theta), st = sinf(theta);
  const float ia2 = 1.f / (ea * ea), ib2 = 1.f / (eb * eb);
  const float A11 = ct * ct * ia2 + st * st * ib2;
  const float A22 = st * st * ia2 + ct * ct * ib2;
  const float A12 = ct * st * (ia2 - ib2);
  const float ax = A11 * cx + A12 * cy;
  const float ay = A12 * cx + A22 * cy;
  const float cAc = A11 * cx * cx + 2.f * A12 * cx * cy + A22 * cy * cy;
  float Cm[3][3] = {{A11, A12, -ax}, {A12, A22, -ay}, {-ax, -ay, cAc - 1.f}};
  const float* Kb = Kmat + b * 9;
  float T[3][3], Cn[3][3];
  for (int i = 0; i < 3; ++i)
    for (int l = 0; l < 3; ++l) {
      float s = 0.f;
      for (int k = 0; k < 3; ++k) s += Cm[i][k] * Kb[k * 3 + l];
      T[i][l] = s;
    }
  for (int i = 0; i < 3; ++i)
    for (int l = 0; l < 3; ++l) {
      float s = 0.f;
      for (int j = 0; j < 3; ++j) s += Kb[j * 3 + i] * T[j][l];
      Cn[i][l] = s;
    }
  // mu = solve(A2, -u), A2 = Cn[:2,:2], u = Cn[:2,2]
  const float a2 = Cn[0][0], b2 = Cn[0][1], c2 = Cn[1][0], d2 = Cn[1][1];
  const float u0 = Cn[0][2], u1 = Cn[1][2];
  const float det = a2 * d2 - b2 * c2;
  const float mu0 = (d2 * (-u0) - b2 * (-u1)) / det;
  const float mu1 = (a2 * (-u1) - c2 * (-u0)) / det;
  // symmetric 2x2 eigendecomposition (ascending eigenvalues)
  const float bm = 0.5f * (b2 + c2);
  const float mean = 0.5f * (a2 + d2);
  const float rr = sqrtf(0.25f * (a2 - d2) * (a2 - d2) + bm * bm);
  const float lam0 = mean - rr, lam1 = mean + rr;
  float vx = bm, vy = lam0 - a2;  // eigenvector of lam0
  if (fabsf(vx) + fabsf(vy) < 1e-20f) {
    vx = (a2 <= d2) ? 1.f : 0.f;
    vy = (a2 <= d2) ? 0.f : 1.f;
  }
  const float theta_n = atan2f(vy, vx);
  float a_n = 1.f / sqrtf(fmaxf(lam0, 1e-12f));
  float b_n = 1.f / sqrtf(fmaxf(lam1, 1e-12f));
  a_n = fmaxf(a_n, 1e-6f);
  const float R = iris_r[b];
  const float z = fminf(fmaxf(R / a_n, 0.5f), 1000.0f);
  const float rl = sqrtf(mu0 * mu0 + mu1 * mu1 + 1.f) + 1e-8f;
  const float ic0 = (mu0 / rl) * z, ic1 = (mu1 / rl) * z, ic2 = (1.f / rl) * z;
  const float ratio = fminf(fmaxf(b_n / a_n, 0.f), 1.f);
  const float tilt = acosf(ratio);
  const float ctl = cosf(tilt), stl = sinf(tilt);
  const float ck = cosf(theta_n), sk = sinf(theta_n);
  const float klen = sqrtf(ck * ck + sk * sk) + 1e-8f;
  const float k0 = ck / klen, k1 = sk / klen;
  // normal = safe_norm(z_ax*ctl + cross(k,z_ax)*stl)  (k.z == 0)
  float n0 = k1 * stl, n1 = -k0 * stl, n2 = ctl;
  const float nl = sqrtf(n0 * n0 + n1 * n1 + n2 * n2) + 1e-8f;
  n0 /= nl; n1 /= nl; n2 /= nl;
  out[448 + be * 3 + 0] = ic0;
  out[448 + be * 3 + 1] = ic1;
  out[448 + be * 3 + 2] = ic2;
  out[640 + be * 3 + 0] = n0;
  out[640 + be * 3 + 1] = n1;
  out[640 + be * 3 + 2] = n2;
  out[832 + be * 3 + 0] = ic0 + delta * n0;
  out[832 + be * 3 + 1] = ic1 + delta * n1;
  out[832 + be * 3 + 2] = ic2 + delta * n2;
  out[1024 + be] = z;
}

// ---------------------------------------------------------------------------
extern "C" void kernel_launch(void* const* d_in, const int* in_sizes, int n_in,
                              void* d_out, int out_size, void* d_ws,
                              size_t ws_size, hipStream_t stream) {
  (void)in_sizes; (void)n_in; (void)out_size; (void)ws_size;
  const float* x        = (const float*)d_in[0];
  const float* Kmat     = (const float*)d_in[1];
  const float* iris     = (const float*)d_in[2];
  const float* ca_w1    = (const float*)d_in[3];
  const float* ca_g     = (const float*)d_in[4];
  const float* ca_b     = (const float*)d_in[5];
  const float* ca_m     = (const float*)d_in[6];
  const float* ca_v     = (const float*)d_in[7];
  const float* ca_wh    = (const float*)d_in[8];
  const float* ca_ww    = (const float*)d_in[9];
  const float* fc1_w    = (const float*)d_in[10];
  const float* fc1_b    = (const float*)d_in[11];
  const float* bn1_g    = (const float*)d_in[12];
  const float* bn1_b    = (const float*)d_in[13];
  const float* bn1_m    = (const float*)d_in[14];
  const float* bn1_v    = (const float*)d_in[15];
  const float* fc_out_w = (const float*)d_in[16];
  const float* fc_out_b = (const float*)d_in[17];

  float* ws  = (float*)d_ws;
  float* xh  = ws;                 // B*C*64
  float* xw  = xh  + B_ * C_ * 64; // B*C*64
  float* a_h = xw  + B_ * C_ * 64; // B*C*64
  float* a_w = a_h + B_ * C_ * 64; // B*C*64
  float* p   = a_w + B_ * C_ * 64; // B*C
  float* hdd = p   + B_ * C_;      // B*HID
  float* out = (float*)d_out;

  pool_kernel<<<B_ * C_, 256, 0, stream>>>(x, xh, xw);
  attn_kernel<<<B_, 256, 0, stream>>>(xh, xw, ca_w1, ca_g, ca_b, ca_m, ca_v,
                                      ca_wh, ca_ww, a_h, a_w);
  wpool_kernel<<<B_ * C_, 256, 0, stream>>>(x, a_h, a_w, p);
  fc1_kernel<<<1, 256, 0, stream>>>(p, fc1_w, fc1_b, bn1_g, bn1_b, bn1_m,
                                    bn1_v, hdd);
  head_kernel<<<1, 64, 0, stream>>>(hdd, fc_out_w, fc_out_b, Kmat, iris, out);
}